// PRODBSplineLayerMultiFeature_46256797778300
// MI455X (gfx1250) — compile-verified
//
#include <hip/hip_runtime.h>
#include <stdint.h>

#define NKNOTS 64
#define FEAT   128
#define BLOCK  256
#define EPB    2048   // elements per block -> 8 per thread

// Cubic B-spline layer:
//   out[b,f] = sum_{r=0..3} Bspline3_{j+r}(x[b,f]) * control_p[j+r, f] + bias[f]
// where j = floor(61*x) (clamped) selects the 4 nonzero cubic basis funcs.
__global__ __launch_bounds__(BLOCK)
void bspline_layer_kernel(const float* __restrict__ x,
                          const float* __restrict__ control_p,
                          const float* __restrict__ bias,
                          float* __restrict__ out,
                          int total)
{
    __shared__ __align__(16) float s_cp[NKNOTS * FEAT];  // 32 KB
    __shared__ __align__(16) float s_bias[FEAT];         // 512 B

    const int tid = threadIdx.x;

    // ---- Stage control_p (+bias) into LDS via CDNA5 async-to-LDS DMA ----
    {
        const float* src = control_p + tid * 4;
        uint32_t dst = (uint32_t)(uintptr_t)(&s_cp[tid * 4]);   // low 32b of flat addr = LDS offset
#pragma unroll
        for (int i = 0; i < 8; ++i) {                            // 256 thr * 8 * 16B = 32 KB
            asm volatile("global_load_async_to_lds_b128 %0, %1, off"
                         :: "v"(dst), "v"(src) : "memory");
            src += BLOCK * 4;
            dst += BLOCK * 16;
        }
        if (tid < 32) {
            const float* bsrc = bias + tid * 4;
            uint32_t bdst = (uint32_t)(uintptr_t)(&s_bias[tid * 4]);
            asm volatile("global_load_async_to_lds_b128 %0, %1, off"
                         :: "v"(bdst), "v"(bsrc) : "memory");
        }
        asm volatile("s_wait_asynccnt 0" ::: "memory");
    }
    __syncthreads();

    const float h = 1.0f / 61.0f;
    const int base = blockIdx.x * EPB + tid;

#pragma unroll 4
    for (int it = 0; it < EPB / BLOCK; ++it) {
        const int idx = base + it * BLOCK;
        if (idx < total) {
            const float xv = x[idx];
            const int f = idx & (FEAT - 1);

            // knot interval: t[m] <= x < t[m+1], m = j+3, j in [0,60]
            int j = (int)(xv * 61.0f);
            j = (j < 60) ? j : 60;
            const float fj = (float)j;

            // knots t[i] = clamp((i-3)*h, 0, 1); indices m-2..m+3
            const float tm   = fj * h;                          // t[m]
            const float tm1  = (fj + 1.0f) * h;                 // t[m+1]
            const float tm2  = fminf((fj + 2.0f) * h, 1.0f);    // t[m+2]
            const float tm3  = fminf((fj + 3.0f) * h, 1.0f);    // t[m+3]
            const float tmm1 = fmaxf((fj - 1.0f) * h, 0.0f);    // t[m-1]
            const float tmm2 = fmaxf((fj - 2.0f) * h, 0.0f);    // t[m-2]

            // local de Boor (denominators never zero for a valid interval)
            // k = 1   (t[m+1]-t[m] == h exactly by construction)
            float r  = 61.0f;
            float N0 = (tm1 - xv) * r;
            float N1 = (xv - tm) * r;
            // k = 2
            r = N0 * __builtin_amdgcn_rcpf(tm1 - tmm1);
            float M0 = (tm1 - xv) * r;
            float sv = (xv - tmm1) * r;
            r = N1 * __builtin_amdgcn_rcpf(tm2 - tm);
            float M1 = sv + (tm2 - xv) * r;
            float M2 = (xv - tm) * r;
            // k = 3
            r = M0 * __builtin_amdgcn_rcpf(tm1 - tmm2);
            float B0 = (tm1 - xv) * r;
            sv = (xv - tmm2) * r;
            r = M1 * __builtin_amdgcn_rcpf(tm2 - tmm1);
            float B1 = sv + (tm2 - xv) * r;
            sv = (xv - tmm1) * r;
            r = M2 * __builtin_amdgcn_rcpf(tm3 - tm);
            float B2 = sv + (tm3 - xv) * r;
            float B3 = (xv - tm) * r;

            // gather 4 control rows from LDS (conflict-free: bank = f mod 64)
            const float* cp = &s_cp[j * FEAT + f];
            float acc = s_bias[f];
            acc = fmaf(B0, cp[0 * FEAT], acc);
            acc = fmaf(B1, cp[1 * FEAT], acc);
            acc = fmaf(B2, cp[2 * FEAT], acc);
            acc = fmaf(B3, cp[3 * FEAT], acc);

            out[idx] = acc;
        }
    }
}

extern "C" void kernel_launch(void* const* d_in, const int* in_sizes, int n_in,
                              void* d_out, int out_size, void* d_ws, size_t ws_size,
                              hipStream_t stream) {
    const float* x    = (const float*)d_in[0];   // [16384, 128]
    const float* cp   = (const float*)d_in[1];   // [64, 128]
    const float* bias = (const float*)d_in[2];   // [128]
    float* out = (float*)d_out;

    const int total = in_sizes[0];               // 2,097,152
    const int blocks = (total + EPB - 1) / EPB;  // 1024

    hipLaunchKernelGGL(bspline_layer_kernel, dim3(blocks), dim3(BLOCK), 0, stream,
                       x, cp, bias, out, total);
}